// LineLoss_17858474017092
// MI455X (gfx1250) — compile-verified
//
#include <hip/hip_runtime.h>
#include <math.h>

#define Bb 32
#define Ll 512
#define Hh 352
#define Ww 1216
// MAXLEN = 1216, but we only walk i = 0..dmaj (beyond that diffs are 0 by clamping).

typedef __attribute__((ext_vector_type(2))) float v2f;
typedef __attribute__((ext_vector_type(8))) float v8f;

__device__ __forceinline__ int sgn_i(int v) { return (v > 0) - (v < 0); }

// ---------------------------------------------------------------------------
// Kernel A: one wave32 per line. Lanes cover 32 consecutive Bresenham steps.
// Single gather per path point; neighbor diff via shfl_down (ds_bpermute);
// cross-window diff added by lane 0 using its own v plus the lane-31 carry.
// Only 2 DS crossbar ops per window. The i_eff clamp keeps all 32 lanes
// active forever (padded diffs are exactly 0), so shuffles are always safe.
// The Bresenham floor-division is strength-reduced to a 64-bit magic multiply
// (m = ceil(2^34/den) is wave-uniform; exact since num*e < 2^34 for
// num <= 2*1215*1215+1215 and e < den <= 2432).
// ---------------------------------------------------------------------------
__global__ __launch_bounds__(256) void line_diff_kernel(
    const float* __restrict__ pred, const int* __restrict__ lines,
    float* __restrict__ deg, float* __restrict__ dsum) {
  const int wave = threadIdx.x >> 5;
  const int lane = threadIdx.x & 31;
  const int line = blockIdx.x * 8 + wave;   // 0 .. 16383
  const int b = line >> 9;                  // / 512

  // lines[line] = {x0, y0, x1, y1} -> one global_load_b128 per wave
  const int4 lp = ((const int4*)lines)[line];
  const int x0 = min(max(lp.x, 0), Ww - 1);
  const int y0 = min(max(lp.y, 0), Hh - 1);
  const int x1 = min(max(lp.z, 0), Ww - 1);
  const int y1 = min(max(lp.w, 0), Hh - 1);

  const int dx = abs(x1 - x0), dy = abs(y1 - y0);
  const bool xmaj = dx > dy;
  const int dmaj = xmaj ? dx : dy;
  const unsigned dminu = (unsigned)(xmaj ? dy : dx);
  const int smaj = xmaj ? sgn_i(x1 - x0) : sgn_i(y1 - y0);
  const int smin = xmaj ? sgn_i(y1 - y0) : sgn_i(x1 - x0);
  const int maj0 = xmaj ? x0 : y0;
  const int min0 = xmaj ? y0 : x0;
  const unsigned den = 2u * (unsigned)max(dmaj, 1);
  // m = ceil(2^34 / den); den >= 2 so m fits in 34 bits (u64 ok)
  const unsigned long long m = 0x3FFFFFFFFULL / den + 1ULL;

  const float* __restrict__ p = pred + (size_t)b * (Hh * Ww);

  float acc = 0.0f;       // per-lane partial of sum |v[j+1]-v[j]|
  float prev_last = 0.0f; // v[base-1] carry (consumed by lane 0 only)
  float vfirst = 0.0f;    // lane 0 keeps v[0] privately

  for (int base = 0; base <= dmaj; base += 32) {
    const unsigned ieff = (unsigned)min(base + lane, dmaj);
    const unsigned num = 2u * dminu * ieff + (unsigned)dmaj;
    const int q = (int)(unsigned)(((unsigned long long)num * m) >> 34);
    const int maj = maj0 + smaj * (int)ieff;
    const int mnr = min0 + smin * q;
    const int xx = xmaj ? maj : mnr;
    const int yy = xmaj ? mnr : maj;
    const float v = __ldg(p + yy * Ww + xx);

    const float nxt = __shfl_down(v, 1);
    if (lane < 31) acc += fabsf(nxt - v);       // diff j = base+lane (auto-0 past dmaj)
    if (base == 0) vfirst = v;                  // lane0's copy == v[0]
    else if (lane == 0) acc += fabsf(v - prev_last); // cross-window diff j = base-1
    prev_last = __shfl(v, 31);                  // after last window this is v[dmaj]
  }

  // wave32 butterfly sum (all lanes end with the total)
  for (int off = 16; off; off >>= 1) acc += __shfl_xor(acc, off);

  if (lane == 0) {
    const float end_diff = fabsf(vfirst - prev_last); // v[0] - v[dmaj]
    const float ds = (end_diff < 0.5f) ? 0.0f : acc;
    // angle from clipped coords, same op order as reference
    const float sl = fabsf(((float)y1 - (float)y0) / ((float)x1 - (float)x0 + 1e-6f));
    const float dg = fmodf(atanf(sl) * 180.0f / 3.14159265358979323846f, 90.0f);
    deg[line]  = dg;
    dsum[line] = ds;
  }
}

// ---------------------------------------------------------------------------
// Kernel B: one workgroup per batch b. LDS tree-reduce max(degrees), then
// angle weights * diff_sums, LDS tree-reduce sum -> part[b].
// ---------------------------------------------------------------------------
__global__ __launch_bounds__(512) void batch_weight_kernel(
    const float* __restrict__ deg, const float* __restrict__ dsum,
    float* __restrict__ part) {
  __shared__ float sh[512];
  const int b = blockIdx.x;
  const int t = threadIdx.x;

  const float d = deg[b * Ll + t];
  sh[t] = d;
  __syncthreads();
  for (int s = 256; s > 0; s >>= 1) {
    if (t < s) sh[t] = fmaxf(sh[t], sh[t + s]);
    __syncthreads();
  }
  const float mx = sh[0];
  __syncthreads();

  const float skewed = powf(d, 1.5f) / powf(mx + 1.0f, 0.5f);
  const float h = 90.0f - powf(powf(fabsf(skewed - 45.0f), 2.5f), 0.4f);
  const float w = h * (1.0f / 300.0f);
  const float w2 = w * w;
  const float wt = w2 * w2;                    // (h/300)^4 via integer_pow expansion
  const float prod = wt * dsum[b * Ll + t];

  sh[t] = prod;
  __syncthreads();
  for (int s = 256; s > 0; s >>= 1) {
    if (t < s) sh[t] += sh[t + s];
    __syncthreads();
  }
  if (t == 0) part[b] = sh[0];
}

// ---------------------------------------------------------------------------
// Kernel C: final reduction of the 32 batch partials with one
// v_wmma_f32_16x16x4_f32: A[m][k] = part[4m+k] (m<8), B = ones(4x16),
// D[m][n] = sum_k A[m][k] -> lane 0 holds rows M=0..7 in its 8 C VGPRs,
// so sum of lane0's accumulators == sum(part[0..31]) exactly (f32 adds).
// EXEC is all-ones at the WMMA (masked loads restore exec before it).
// ---------------------------------------------------------------------------
__global__ __launch_bounds__(32) void final_reduce_wmma(
    const float* __restrict__ part, float* __restrict__ out) {
  const int lane = threadIdx.x;       // 0..31
  const int M = lane & 15;
  const int kb = (lane >> 4) * 2;     // lanes 0-15: K=0,1 ; lanes 16-31: K=2,3

  v2f a;
  a.x = (M < 8) ? part[M * 4 + kb + 0] : 0.0f;
  a.y = (M < 8) ? part[M * 4 + kb + 1] : 0.0f;
  v2f bm;
  bm.x = 1.0f;
  bm.y = 1.0f;
  v8f c = {};
  c = __builtin_amdgcn_wmma_f32_16x16x4_f32(
      /*neg_a=*/false, a, /*neg_b=*/false, bm,
      /*c_mod=*/(short)0, c, /*reuse_a=*/false, /*reuse_b=*/false);

  float s = c[0] + c[1] + c[2] + c[3] + c[4] + c[5] + c[6] + c[7];
  const float total = __shfl(s, 0) + __shfl(s, 16); // lanes 16+ rows are all zero
  if (lane == 0) out[0] = total;
}

// ---------------------------------------------------------------------------
extern "C" void kernel_launch(void* const* d_in, const int* in_sizes, int n_in,
                              void* d_out, int out_size, void* d_ws, size_t ws_size,
                              hipStream_t stream) {
  const float* pred = (const float*)d_in[0];   // [32,1,352,1216] f32
  const int* lines  = (const int*)d_in[1];     // [32,512,2,2] i32
  float* out = (float*)d_out;                  // scalar f32

  float* deg  = (float*)d_ws;                  // 16384 floats
  float* dsum = deg + Bb * Ll;                 // 16384 floats
  float* part = dsum + Bb * Ll;                // 32 floats

  // 16384 lines, 8 waves (256 threads) per block -> 2048 blocks
  line_diff_kernel<<<(Bb * Ll) / 8, 256, 0, stream>>>(pred, lines, deg, dsum);
  batch_weight_kernel<<<Bb, Ll, 0, stream>>>(deg, dsum, part);
  final_reduce_wmma<<<1, 32, 0, stream>>>(part, out);
}